// MultiSpectralChannelAttention_68143951118912
// MI455X (gfx1250) — compile-verified
//
#include <hip/hip_runtime.h>
#include <math.h>

typedef float v2f __attribute__((ext_vector_type(2)));
typedef float v8f __attribute__((ext_vector_type(8)));
typedef int   v4i __attribute__((ext_vector_type(4)));

// Problem dims (fixed by the reference)
#define B_   32
#define C_   256
#define F_   16
#define HW_  4096              // 64*64
#define CHW_ (C_ * HW_)        // 1048576
#define R_   16                // reduction ratio -> hidden = 16

#define CHUNK 128              // K-chunk staged in LDS
#define LDA   132              // padded LDS row stride (floats) -> bank-conflict-free column reads
#define LDB   132

// ---- gfx1250 async global->LDS path (guarded; falls back to VGPR staging) --
#if defined(__has_builtin)
#  if __has_builtin(__builtin_amdgcn_global_load_async_to_lds_b128)
#    define HAVE_ASYNC_LDS 1
#  endif
#  if __has_builtin(__builtin_amdgcn_s_wait_asynccnt)
#    define HAVE_WAIT_ASYNC_BUILTIN 1
#  endif
#endif

#if defined(HAVE_ASYNC_LDS)
typedef __attribute__((address_space(1))) v4i* as1_v4i_ptr;   // global
typedef __attribute__((address_space(3))) v4i* as3_v4i_ptr;   // LDS
#endif

__device__ __forceinline__ void async_wait0() {
#if defined(HAVE_ASYNC_LDS)
#  if defined(HAVE_WAIT_ASYNC_BUILTIN)
    __builtin_amdgcn_s_wait_asynccnt(0);
#  else
    asm volatile("s_wait_asynccnt 0x0" ::: "memory");
#  endif
#endif
}

__device__ __forceinline__ void cp16_to_lds(const float* __restrict__ g, float* l) {
#if defined(HAVE_ASYNC_LDS)
    // global_load_async_to_lds_b128: direct HBM->LDS, tracked by ASYNCcnt
    __builtin_amdgcn_global_load_async_to_lds_b128(
        (as1_v4i_ptr)(g), (as3_v4i_ptr)(l), 0, 0);
#else
    *(float4*)l = *(const float4*)g;
#endif
}

// Stage one 128-wide K chunk: x tile 32x128 (1024 float4, 4/thread) and
// dct tile 16x128 (512 float4, 2/thread).  Fixed trip counts -> uniform flow.
__device__ __forceinline__ void stage_tiles(const float* __restrict__ xb,
                                            const float* __restrict__ db,
                                            float* lx, float* ld,
                                            int k0, int tid) {
    #pragma unroll
    for (int it = 0; it < 4; ++it) {
        const int i = tid + it * 256;
        const int r = i >> 5;                 // 32 float4 per row
        const int q = (i & 31) << 2;          // float column
        cp16_to_lds(xb + (size_t)r * CHW_ + k0 + q, &lx[r * LDA + q]);
    }
    #pragma unroll
    for (int it = 0; it < 2; ++it) {
        const int i = tid + it * 256;
        const int r = i >> 5;
        const int q = (i & 31) << 2;
        cp16_to_lds(db + (size_t)r * HW_ + k0 + q, &ld[r * LDB + q]);
    }
}

// ---------------------------------------------------------------------------
// K1: per-channel GEMM  S_c = X_c (32 x 4096) @ D_c^T (4096 x 16) via
//     v_wmma_f32_16x16x4_f32, double-buffered async HBM->LDS staging.
//     One block per channel, 8 waves: waves 0-3 -> M-tile 0 (b=0..15),
//     waves 4-7 -> M-tile 1 (b=16..31); each wave owns a 32-wide K sub-slice.
// Output feats[b][f*256 + c]  (reference's (b, f*c) reshape order).
// ---------------------------------------------------------------------------
__global__ __launch_bounds__(256)
void msca_dct_gemm_kernel(const float* __restrict__ x,
                          const float* __restrict__ dct_w,
                          float* __restrict__ feats) {
    const int cch = blockIdx.x;            // channel
    const int tid = threadIdx.x;
    const int lane = tid & 31;
    const int wid  = tid >> 5;
    const int mt   = wid >> 2;             // 0..1  (M tile)
    const int ks   = wid & 3;              // 0..3  (K sub-slice)

    __shared__ __align__(16) float ldsX[2][B_ * LDA];  // 2 x (32 rows x 128+pad)
    __shared__ __align__(16) float ldsD[2][F_ * LDB];  // 2 x (16 rows x 128+pad)
    __shared__ float red[2][3][256];                   // cross-wave K partials

    const int m     = lane & 15;           // row-within-tile / N index
    const int khalf = (lane >> 4) << 1;    // 0 for lanes 0-15, 2 for 16-31

    v8f acc = {0.f, 0.f, 0.f, 0.f, 0.f, 0.f, 0.f, 0.f};

    const float* xb = x + (size_t)cch * HW_;
    const float* db = dct_w + (size_t)cch * F_ * HW_;

    // prologue: fill buffer 0
    stage_tiles(xb, db, ldsX[0], ldsD[0], 0, tid);
    async_wait0();
    __syncthreads();

    for (int k0 = 0; k0 < HW_; k0 += CHUNK) {
        const int cur = (k0 >> 7) & 1;
        const int nxt = cur ^ 1;

        // overlap: kick off next chunk's HBM->LDS fill (uniform branch)
        if (k0 + CHUNK < HW_)
            stage_tiles(xb, db, ldsX[nxt], ldsD[nxt], k0 + CHUNK, tid);

        // WMMA over this wave's 32-wide K slice (8 steps of K=4)
        const float* ax = &ldsX[cur][(mt * 16 + m) * LDA + ks * 32 + khalf];
        const float* bx = &ldsD[cur][m * LDB + ks * 32 + khalf];
        #pragma unroll
        for (int kk = 0; kk < 32; kk += 4) {
            v2f a = *(const v2f*)(ax + kk);   // A: X[b=mtile+m][k]  (ds_load_b64)
            v2f b = *(const v2f*)(bx + kk);   // B: dct_w[c][n=m][k]
            acc = __builtin_amdgcn_wmma_f32_16x16x4_f32(
                      false, a, false, b, (short)0, acc, false, false);
        }

        async_wait0();     // next buffer's async fills have landed
        __syncthreads();   // ... in every wave
    }

    // ---- reduce the 4 K-partials per M tile ----
    if (ks != 0) {
        #pragma unroll
        for (int j = 0; j < 8; ++j) red[mt][ks - 1][lane * 8 + j] = acc[j];
    }
    __syncthreads();
    if (ks == 0) {
        #pragma unroll
        for (int j = 0; j < 8; ++j) {
            float v = acc[j] + red[mt][0][lane * 8 + j]
                             + red[mt][1][lane * 8 + j]
                             + red[mt][2][lane * 8 + j];
            // C/D layout: VGPR j = rows M=j (lanes 0-15) and M=j+8 (lanes 16-31)
            const int brow = mt * 16 + ((lane < 16) ? j : j + 8);
            const int fcol = lane & 15;
            feats[(size_t)brow * (F_ * C_) + fcol * C_ + cch] = v;
        }
    }
}

// ---------------------------------------------------------------------------
// K2: per-batch-row MLP.  block b: hidden[j] = relu(feats[b]·W1[j]) (16 dots
// of length 4096, block-reduced), then att[b][c] = sigmoid(hidden·W2[c]).
// ---------------------------------------------------------------------------
__global__ __launch_bounds__(256)
void msca_mlp_kernel(const float* __restrict__ feats,
                     const float* __restrict__ W1,
                     const float* __restrict__ W2,
                     float* __restrict__ att) {
    const int b   = blockIdx.x;
    const int tid = threadIdx.x;
    __shared__ float red[256];
    __shared__ float hidden[R_];

    const float* frow = feats + (size_t)b * (F_ * C_);
    for (int j = 0; j < R_; ++j) {
        const float* wrow = W1 + (size_t)j * (F_ * C_);
        float p = 0.f;
        for (int k = tid; k < F_ * C_; k += 256) p = fmaf(frow[k], wrow[k], p);
        red[tid] = p;
        __syncthreads();
        for (int s = 128; s > 0; s >>= 1) {
            if (tid < s) red[tid] += red[tid + s];
            __syncthreads();
        }
        if (tid == 0) hidden[j] = fmaxf(red[0], 0.f);
        __syncthreads();
    }

    // att for channel tid
    const float* w2r = W2 + tid * R_;
    float s = 0.f;
    #pragma unroll
    for (int j = 0; j < R_; ++j) s = fmaf(hidden[j], w2r[j], s);
    att[b * C_ + tid] = 1.f / (1.f + expf(-s));
}

// ---------------------------------------------------------------------------
// K3: out = x * att[b,c], float4 grid-stride (HW=4096 is a multiple of 4, so
// all 4 elements of a float4 share one att value).
// ---------------------------------------------------------------------------
__global__ __launch_bounds__(256)
void msca_scale_kernel(const float* __restrict__ x,
                       const float* __restrict__ att,
                       float* __restrict__ out,
                       long n4) {
    long i = (long)blockIdx.x * 256 + threadIdx.x;
    const long stride = (long)gridDim.x * 256;
    for (; i < n4; i += stride) {
        float4 v = ((const float4*)x)[i];
        const int bc = (int)((i << 2) >> 12);     // (element index) / 4096
        const float a = att[bc];
        v.x *= a; v.y *= a; v.z *= a; v.w *= a;
        ((float4*)out)[i] = v;
    }
}

// ---------------------------------------------------------------------------
extern "C" void kernel_launch(void* const* d_in, const int* in_sizes, int n_in,
                              void* d_out, int out_size, void* d_ws, size_t ws_size,
                              hipStream_t stream) {
    const float* x     = (const float*)d_in[0];   // (32,256,64,64)
    const float* dct_w = (const float*)d_in[1];   // (256,16,64,64)
    const float* W1    = (const float*)d_in[2];   // (16,4096)
    const float* W2    = (const float*)d_in[3];   // (256,16)
    float* out = (float*)d_out;

    // workspace: feats (32*4096 f32 = 512KB) then att (32*256 f32 = 32KB)
    float* feats = (float*)d_ws;
    float* att   = (float*)((char*)d_ws + (size_t)B_ * F_ * C_ * sizeof(float));

    msca_dct_gemm_kernel<<<dim3(C_), dim3(256), 0, stream>>>(x, dct_w, feats);
    msca_mlp_kernel<<<dim3(B_), dim3(256), 0, stream>>>(feats, W1, W2, att);

    const long n4 = (long)B_ * C_ * HW_ / 4;      // 8,388,608 float4
    msca_scale_kernel<<<dim3(8192), dim3(256), 0, stream>>>(x, att, out, n4);
}